// DynMoE_53747220742504
// MI455X (gfx1250) — compile-verified
//
#include <hip/hip_runtime.h>
#include <math.h>

// Problem sizes (fixed by the reference).
constexpr int Tn = 2048;        // B*S tokens
constexpr int Hn = 1024;        // hidden
constexpr int En = 8;           // experts
constexpr int In = 512;         // intermediate
constexpr int EI = En * In;     // 4096 (flattened expert*intermediate K for down proj)

typedef float v2f __attribute__((ext_vector_type(2)));
typedef float v8f __attribute__((ext_vector_type(8)));
typedef int   v4i __attribute__((ext_vector_type(4)));

#define GLOBAL_AS __attribute__((address_space(1)))
#define LDS_AS    __attribute__((address_space(3)))

// V_WMMA_F32_16X16X4_F32 : D(16x16 f32) = A(16x4) * B(4x16) + C, exact fp32 path.
__device__ __forceinline__ v8f wmma4(v2f a, v2f b, v8f c) {
  return __builtin_amdgcn_wmma_f32_16x16x4_f32(
      /*neg_a=*/false, a, /*neg_b=*/false, b,
      /*c_mod=*/(short)0, c, /*reuse_a=*/false, /*reuse_b=*/false);
}

// CDNA5 async direct-to-LDS copies (ASYNCcnt-tracked, no VGPR round trip).
__device__ __forceinline__ void async_b32(const float* g, float* l) {
  __builtin_amdgcn_global_load_async_to_lds_b32(
      (GLOBAL_AS int*)g, (LDS_AS int*)l, 0, 0);
}
__device__ __forceinline__ void async_b128(const float* g, float* l) {
  __builtin_amdgcn_global_load_async_to_lds_b128(
      (GLOBAL_AS v4i*)g, (LDS_AS v4i*)l, 0, 0);
}
__device__ __forceinline__ void wait_async0() {
  __builtin_amdgcn_s_wait_asynccnt(0);
}

// ---------------- kernel 0: normalize expert embeddings ----------------
__global__ __launch_bounds__(256) void k_norm_embed(const float* __restrict__ emb,
                                                    float* __restrict__ en) {
  __shared__ float red[256];
  const int e = blockIdx.x;
  const int tid = threadIdx.x;
  float s = 0.f;
  for (int h = tid; h < Hn; h += 256) { float v = emb[e * Hn + h]; s += v * v; }
  red[tid] = s;
  __syncthreads();
  for (int off = 128; off > 0; off >>= 1) {
    if (tid < off) red[tid] += red[tid + off];
    __syncthreads();
  }
  const float inv = 1.f / fmaxf(sqrtf(red[0]), 1e-8f);
  for (int h = tid; h < Hn; h += 256) en[e * Hn + h] = emb[e * Hn + h] * inv;
}

// ---------------- kernel 1: routing (wave per token) ----------------
__global__ __launch_bounds__(256) void k_route(const float* __restrict__ X,
                                               const float* __restrict__ en,
                                               const float* __restrict__ thr,
                                               float* __restrict__ logits,
                                               float* __restrict__ gates) {
  const int lane = threadIdx.x & 31;
  const int wid  = threadIdx.x >> 5;
  const int t = blockIdx.x * 8 + wid;
  const float* x = X + (size_t)t * Hn;

  float dot[En];
#pragma unroll
  for (int e = 0; e < En; ++e) dot[e] = 0.f;
  float sq = 0.f;
  for (int h = lane; h < Hn; h += 32) {
    const float xv = x[h];
    sq += xv * xv;
#pragma unroll
    for (int e = 0; e < En; ++e) dot[e] += xv * en[e * Hn + h];
  }
#pragma unroll
  for (int off = 16; off > 0; off >>= 1) {
    sq += __shfl_xor(sq, off, 32);
#pragma unroll
    for (int e = 0; e < En; ++e) dot[e] += __shfl_xor(dot[e], off, 32);
  }
  if (lane == 0) {
    const float inv = 1.f / fmaxf(sqrtf(sq), 1e-8f);
    float sc[En];
    float mx = -INFINITY; int am = 0;
#pragma unroll
    for (int e = 0; e < En; ++e) {
      sc[e] = dot[e] * inv;
      logits[t * En + e] = sc[e];
      if (sc[e] > mx) { mx = sc[e]; am = e; }   // first-max, matches jnp.argmax
    }
    // silu(score - thr) > 0  <=>  score > thr
    int cnt = 0; float smax = -INFINITY;
    bool m[En];
#pragma unroll
    for (int e = 0; e < En; ++e) {
      m[e] = sc[e] > thr[e];
      if (m[e]) { cnt++; if (sc[e] > smax) smax = sc[e]; }
    }
    float w[En]; float sum = 0.f;
#pragma unroll
    for (int e = 0; e < En; ++e) {
      w[e] = m[e] ? expf(sc[e] - smax) : 0.f;
      sum += w[e];
    }
    const float rs = 1.f / sum;
#pragma unroll
    for (int e = 0; e < En; ++e) {
      const bool fin = cnt ? m[e] : (e == am);
      gates[t * En + e] = fin ? w[e] * rs : 0.f;
    }
  }
}

// ---------------- fp32 WMMA GEMM tiling ----------------
constexpr int BM  = 128;   // token tile
constexpr int BN  = 128;   // col tile
constexpr int BK  = 16;    // k tile
constexpr int BKP = 20;    // padded A-tile LDS row stride (bank-conflict-free b64 frags)
// B tiles use a k-paired layout: element (k,n) at [(k>>1)*BPR + 2n + (k&1)] so each
// lane's fragment {B[kb][n], B[kb+1][n]} is one aligned ds_load_b64.
// BPR = 288 floats: +288 ≡ +32 banks (mod 64) separates the half-waves.
constexpr int BPR = 288;                  // floats per k-pair row
constexpr int BTILE = (BK / 2) * BPR;     // 2304 floats per B tile

// kernel 2: per-expert gate+up GEMMs + SwiGLU + routing gate -> hmid[T, E*I]
__global__ __launch_bounds__(256) void k_swiglu(const float* __restrict__ X,
                                                const float* __restrict__ Wg_all,
                                                const float* __restrict__ Wu_all,
                                                const float* __restrict__ gates,
                                                float* __restrict__ hmid) {
  __shared__ __align__(16) float sX[2][BM * BKP];
  __shared__ __align__(16) float sG[2][BTILE];
  __shared__ __align__(16) float sU[2][BTILE];

  const int e = blockIdx.z;
  const float* Wg = Wg_all + (size_t)e * Hn * In;   // [H, I]
  const float* Wu = Wu_all + (size_t)e * Hn * In;

  const int tid  = threadIdx.x;
  const int lane = tid & 31;
  const int wid  = tid >> 5;
  const int wm   = (wid >> 1) * 32;   // wave row offset inside block tile (4 waves down)
  const int wn   = (wid & 1) * 64;    // wave col offset inside block tile (2 waves across)
  const int m0 = blockIdx.y * BM;
  const int n0 = blockIdx.x * BN;
  const int l15   = lane & 15;
  const int khalf = (lane < 16) ? 0 : 2;   // A/B fragment K split per ISA layout

  v8f accG[2][4], accU[2][4];
  const v8f vzero = {0.f, 0.f, 0.f, 0.f, 0.f, 0.f, 0.f, 0.f};
#pragma unroll
  for (int i = 0; i < 2; ++i)
#pragma unroll
    for (int j = 0; j < 4; ++j) { accG[i][j] = vzero; accU[i][j] = vzero; }

  // X tile: 2 async b128 per thread (512 float4's per tile)
  const int xrow = tid >> 2;           // 0..63 (+64 on 2nd pass)
  const int xcol = (tid & 3) * 4;      // 0,4,8,12
  // W tiles: 16 async b32 per thread into paired layout
  const int wnl = tid & 127;           // W-tile col
  const int wkl = tid >> 7;            // W-tile row parity (wave-uniform)

  auto load_tiles = [&](int buf, int k0) {
#pragma unroll
    for (int p = 0; p < 2; ++p) {
      const int row = xrow + p * 64;
      async_b128(&X[(size_t)(m0 + row) * Hn + k0 + xcol],
                 &sX[buf][row * BKP + xcol]);
    }
#pragma unroll
    for (int kk = 0; kk < BK; kk += 2) {
      const int sidx = (kk >> 1) * BPR + 2 * wnl + wkl;
      const size_t gofs = (size_t)(k0 + kk + wkl) * In + n0 + wnl;
      async_b32(&Wg[gofs], &sG[buf][sidx]);
      async_b32(&Wu[gofs], &sU[buf][sidx]);
    }
  };

  load_tiles(0, 0);
  for (int k0 = 0; k0 < Hn; k0 += BK) {
    const int cur = (k0 >> 4) & 1;
    wait_async0();
    __syncthreads();
    if (k0 + BK < Hn) load_tiles(cur ^ 1, k0 + BK);
#pragma unroll
    for (int kk = 0; kk < BK; kk += 4) {
      v2f a[2];
#pragma unroll
      for (int ms = 0; ms < 2; ++ms)
        a[ms] = *(const v2f*)&sX[cur][(wm + ms * 16 + l15) * BKP + kk + khalf];
      const int prow = ((kk + khalf) >> 1) * BPR;
#pragma unroll
      for (int ns = 0; ns < 4; ++ns) {
        const int col2 = 2 * (wn + ns * 16 + l15);
        const v2f bg = *(const v2f*)&sG[cur][prow + col2];
        const v2f bu = *(const v2f*)&sU[cur][prow + col2];
#pragma unroll
        for (int ms = 0; ms < 2; ++ms) {
          accG[ms][ns] = wmma4(a[ms], bg, accG[ms][ns]);
          accU[ms][ns] = wmma4(a[ms], bu, accU[ms][ns]);
        }
      }
    }
  }

  const int rb = (lane < 16) ? 0 : 8;   // C/D layout: VGPR r holds M=r (lo half) / r+8 (hi half)
#pragma unroll
  for (int ms = 0; ms < 2; ++ms)
#pragma unroll
    for (int ns = 0; ns < 4; ++ns)
#pragma unroll
      for (int r = 0; r < 8; ++r) {
        const int row = m0 + wm + ms * 16 + rb + r;
        const int col = n0 + wn + ns * 16 + l15;
        const float g  = accG[ms][ns][r];
        const float u  = accU[ms][ns][r];
        const float sg = g / (1.f + __expf(-g));           // silu
        hmid[(size_t)row * EI + e * In + col] = gates[row * En + e] * sg * u;
      }
}

// kernel 3: Out[T,H] = hmid[T, E*I] @ w_down_flat[E*I, H]   (gates already folded in)
__global__ __launch_bounds__(256) void k_down(const float* __restrict__ A,
                                              const float* __restrict__ B,
                                              float* __restrict__ Out) {
  __shared__ __align__(16) float sA[2][BM * BKP];
  __shared__ __align__(16) float sB[2][BTILE];

  const int tid  = threadIdx.x;
  const int lane = tid & 31;
  const int wid  = tid >> 5;
  const int wm   = (wid >> 1) * 32;
  const int wn   = (wid & 1) * 64;
  const int m0 = blockIdx.y * BM;
  const int n0 = blockIdx.x * BN;
  const int l15   = lane & 15;
  const int khalf = (lane < 16) ? 0 : 2;

  v8f acc[2][4];
  const v8f vzero = {0.f, 0.f, 0.f, 0.f, 0.f, 0.f, 0.f, 0.f};
#pragma unroll
  for (int i = 0; i < 2; ++i)
#pragma unroll
    for (int j = 0; j < 4; ++j) acc[i][j] = vzero;

  const int xrow = tid >> 2;
  const int xcol = (tid & 3) * 4;
  const int bnl = tid & 127;
  const int bkl = tid >> 7;   // wave-uniform parity

  auto load_tiles = [&](int buf, int k0) {
#pragma unroll
    for (int p = 0; p < 2; ++p) {
      const int row = xrow + p * 64;
      async_b128(&A[(size_t)(m0 + row) * EI + k0 + xcol],
                 &sA[buf][row * BKP + xcol]);
    }
#pragma unroll
    for (int kk = 0; kk < BK; kk += 2)
      async_b32(&B[(size_t)(k0 + kk + bkl) * Hn + n0 + bnl],
                &sB[buf][(kk >> 1) * BPR + 2 * bnl + bkl]);
  };

  load_tiles(0, 0);
  for (int k0 = 0; k0 < EI; k0 += BK) {
    const int cur = (k0 >> 4) & 1;
    wait_async0();
    __syncthreads();
    if (k0 + BK < EI) load_tiles(cur ^ 1, k0 + BK);
#pragma unroll
    for (int kk = 0; kk < BK; kk += 4) {
      v2f a[2];
#pragma unroll
      for (int ms = 0; ms < 2; ++ms)
        a[ms] = *(const v2f*)&sA[cur][(wm + ms * 16 + l15) * BKP + kk + khalf];
      const int prow = ((kk + khalf) >> 1) * BPR;
#pragma unroll
      for (int ns = 0; ns < 4; ++ns) {
        const v2f b = *(const v2f*)&sB[cur][prow + 2 * (wn + ns * 16 + l15)];
#pragma unroll
        for (int ms = 0; ms < 2; ++ms) acc[ms][ns] = wmma4(a[ms], b, acc[ms][ns]);
      }
    }
  }

  const int rb = (lane < 16) ? 0 : 8;
#pragma unroll
  for (int ms = 0; ms < 2; ++ms)
#pragma unroll
    for (int ns = 0; ns < 4; ++ns)
#pragma unroll
      for (int r = 0; r < 8; ++r) {
        const int row = m0 + wm + ms * 16 + rb + r;
        const int col = n0 + wn + ns * 16 + l15;
        Out[(size_t)row * Hn + col] = acc[ms][ns][r];
      }
}

// ---------------- host launcher ----------------
extern "C" void kernel_launch(void* const* d_in, const int* in_sizes, int n_in,
                              void* d_out, int out_size, void* d_ws, size_t ws_size,
                              hipStream_t stream) {
  const float* X   = (const float*)d_in[0];   // [T, H]
  const float* emb = (const float*)d_in[1];   // [E, H]
  const float* thr = (const float*)d_in[2];   // [E]
  const float* wg  = (const float*)d_in[3];   // [E, H, I]
  const float* wu  = (const float*)d_in[4];   // [E, H, I]
  const float* wd  = (const float*)d_in[5];   // [E, I, H] == flat [E*I, H]

  float* out    = (float*)d_out;              // [T, H]
  float* logits = out + (size_t)Tn * Hn;      // [T, E]

  // workspace: en | gates | hmid
  float* en    = (float*)d_ws;
  float* gates = en + (size_t)En * Hn;
  float* hmid  = gates + (size_t)Tn * En;     // [T, E*I] ~ 32 MiB

  k_norm_embed<<<En, 256, 0, stream>>>(emb, en);
  k_route<<<Tn / 8, 256, 0, stream>>>(X, en, thr, logits, gates);
  k_swiglu<<<dim3(In / BN, Tn / BM, En), 256, 0, stream>>>(X, wg, wu, gates, hmid);
  k_down<<<dim3(Hn / BN, Tn / BM), 256, 0, stream>>>(hmid, wd, out);
}